// NeuralSpline_30975304139167
// MI455X (gfx1250) — compile-verified
//
#include <hip/hip_runtime.h>

// ---------------------------------------------------------------------------
// Neural spline coupling layer for MI455X (gfx1250, wave32).
//  - Per-pixel 116x4 GEMM done with V_WMMA_F32_16X16X4_F32 (exact shape match)
//  - Spline evaluation done in VALU from an LDS parameter table
// ---------------------------------------------------------------------------

typedef float v2f __attribute__((ext_vector_type(2)));
typedef float v8f __attribute__((ext_vector_type(8)));

#define HW        65536          // 256*256
#define CSTRIDE   65536          // channel stride in floats
#define BSTRIDE   (8 * 65536)    // batch stride in floats (8 channels)
#define NPARAM    116            // 4 channels * 29 params
#define PSTRIDE   117            // odd LDS stride -> conflict-free reads
#define TPB       128            // 4 waves, 128 pixels per block
#define NBLOCKS   4096           // 524288 pixels / 128

__device__ __forceinline__ float softplusf(float v) {
    // numerically stable log(1 + exp(v)) (matches jax.nn.softplus)
    return fmaxf(v, 0.0f) + log1pf(expf(-fabsf(v)));
}

__global__ void zero_lad_kernel(float* __restrict__ lad) {
    if (threadIdx.x < 8) lad[threadIdx.x] = 0.0f;
}

__global__ __launch_bounds__(TPB)
void spline_kernel(const float* __restrict__ x,
                   const float* __restrict__ Wn,
                   const float* __restrict__ bn,
                   float* __restrict__ out,
                   float* __restrict__ lad_out) {
    __shared__ float lds_p[TPB * PSTRIDE];   // per-pixel spline parameters
    __shared__ float lds_bias[NPARAM];
    __shared__ float red[TPB];

    const int t    = threadIdx.x;
    const int lane = t & 31;
    const int wave = t >> 5;
    const int hi16 = lane >> 4;     // 0 for lanes 0-15, 1 for lanes 16-31
    const int l15  = lane & 15;

    if (t < NPARAM) lds_bias[t] = bn[t];
    __syncthreads();

    // ---- A tiles: W_net rows (M) x K=4, ISA layout:
    //      lane M = lane&15 ; VGPR0 holds K = (lane<16 ? 0 : 2), VGPR1 = K+1
    const int k0 = hi16 * 2;
    v2f A[8];
#pragma unroll
    for (int tt = 0; tt < 8; ++tt) {
        int r = tt * 16 + l15;
        v2f a; a.x = 0.0f; a.y = 0.0f;
        if (r < NPARAM) {
            a.x = Wn[r * 4 + k0];
            a.y = Wn[r * 4 + k0 + 1];
        }
        A[tt] = a;
    }

    const int blockPix = blockIdx.x * TPB;
    const int wavePix  = blockPix + wave * 32;

    // ---- GEMM phase: each wave covers 32 pixels = 2 groups of 16 (N dim) ----
#pragma unroll
    for (int g = 0; g < 2; ++g) {
        int pix = wavePix + g * 16 + l15;
        int b   = pix >> 16;            // HW = 65536
        int hw  = pix & (HW - 1);
        const float* xb = x + (size_t)b * BSTRIDE + hw;
        // B matrix 4x16: VGPR0 = rows K0 (lanes 0-15) / K2 (lanes 16-31), VGPR1 = K1/K3
        v2f Bv;
        Bv.x = xb[(size_t)k0 * CSTRIDE];
        Bv.y = xb[(size_t)(k0 + 1) * CSTRIDE];

        int s = wave * 32 + g * 16 + l15;   // LDS pixel slot
#pragma unroll
        for (int tt = 0; tt < 8; ++tt) {
            v8f c = {0.f, 0.f, 0.f, 0.f, 0.f, 0.f, 0.f, 0.f};
            v8f d = __builtin_amdgcn_wmma_f32_16x16x4_f32(
                        false, A[tt], false, Bv, (short)0, c, false, false);
            // D layout: VGPR v holds M = 16*tt + v + (lane<16 ? 0 : 8), N = lane&15
            int mbase = tt * 16 + hi16 * 8;
#pragma unroll
            for (int v = 0; v < 8; ++v) {
                int m = mbase + v;
                if (m < NPARAM)
                    lds_p[s * PSTRIDE + m] = d[v] + lds_bias[m];
            }
        }
    }
    __syncthreads();   // (same-wave LDS is in-order; barrier for cross-wave safety)

    // ---- Spline phase: thread t handles pixel slot t, 4 transform channels ----
    const int pix = blockPix + t;
    const int b   = pix >> 16;
    const int hw  = pix & (HW - 1);
    const float* xb = x   + (size_t)b * BSTRIDE + hw;
    float*       ob = out + (size_t)b * BSTRIDE + hw;

    // identity passthrough (channels 0..3)
#pragma unroll
    for (int i = 0; i < 4; ++i)
        ob[(size_t)i * CSTRIDE] = xb[(size_t)i * CSTRIDE];

    float ladsum = 0.0f;

#pragma unroll 1
    for (int c = 0; c < 4; ++c) {
        const float* Pp = &lds_p[t * PSTRIDE + c * 29];
        float P[29];
#pragma unroll
        for (int j = 0; j < 29; ++j) P[j] = Pp[j];

        const float scale = 0.125f;   // 1/sqrt(64)

        // -------- widths: softmax, MINW floor, cumsum -> knots cw[0..10] ----
        float mw = P[0] * scale;
#pragma unroll
        for (int j = 1; j < 10; ++j) mw = fmaxf(mw, P[j] * scale);
        float ew[10], sw = 0.0f;
#pragma unroll
        for (int j = 0; j < 10; ++j) { ew[j] = expf(P[j] * scale - mw); sw += ew[j]; }
        float invsw = 1.0f / sw;
        float cw[11]; cw[0] = -1.0f;
        float accw = 0.0f;
#pragma unroll
        for (int j = 0; j < 10; ++j) {
            float wj = 0.01f + 0.9f * (ew[j] * invsw);
            accw += wj;
            cw[j + 1] = 2.0f * accw - 1.0f;
        }
        cw[10] = 1.0f;

        // -------- heights ---------------------------------------------------
        float mh = P[10] * scale;
#pragma unroll
        for (int j = 1; j < 10; ++j) mh = fmaxf(mh, P[10 + j] * scale);
        float eh[10], sh = 0.0f;
#pragma unroll
        for (int j = 0; j < 10; ++j) { eh[j] = expf(P[10 + j] * scale - mh); sh += eh[j]; }
        float invsh = 1.0f / sh;
        float chh[11]; chh[0] = -1.0f;
        float acch = 0.0f;
#pragma unroll
        for (int j = 0; j < 10; ++j) {
            float hj = 0.01f + 0.9f * (eh[j] * invsh);
            acch += hj;
            chh[j + 1] = 2.0f * acch - 1.0f;
        }
        chh[10] = 1.0f;

        // -------- derivatives: MIND + softplus; padded endpoints == 1 -------
        float dv[11];
        dv[0] = 1.0f; dv[10] = 1.0f;
#pragma unroll
        for (int j = 1; j < 10; ++j) dv[j] = 0.01f + softplusf(P[19 + j]);

        // -------- input, bin search (static-index select, stays in VGPRs) ---
        float xv = xb[(size_t)(4 + c) * CSTRIDE];
        bool inside = (xv >= -1.0f) && (xv <= 1.0f);
        float xc = fminf(fmaxf(xv, -1.0f), 1.0f);

        int idx = 0;
#pragma unroll
        for (int k = 1; k < 10; ++k)
            if (xc >= cw[k]) idx = k;    // cw[10]+EPS > xc always -> idx <= 9

        float in_cw = cw[0], in_w = cw[1] - cw[0];
        float in_ch = chh[0], in_h = chh[1] - chh[0];
        float in_d = dv[0],  in_dp1 = dv[1];
#pragma unroll
        for (int k = 1; k < 10; ++k) {
            if (idx == k) {
                in_cw = cw[k];  in_w = cw[k + 1] - cw[k];
                in_ch = chh[k]; in_h = chh[k + 1] - chh[k];
                in_d = dv[k];   in_dp1 = dv[k + 1];
            }
        }

        // -------- rational-quadratic spline --------------------------------
        float in_delta = in_h / in_w;
        float theta = (xc - in_cw) / in_w;
        float omt   = 1.0f - theta;
        float t1mt  = theta * omt;
        float th2   = theta * theta;
        float numerator   = in_h * (in_delta * th2 + in_d * t1mt);
        float denominator = in_delta + (in_d + in_dp1 - 2.0f * in_delta) * t1mt;
        float outs = in_ch + numerator / denominator;
        float dnum = in_delta * in_delta *
                     (in_dp1 * th2 + 2.0f * in_delta * t1mt + in_d * omt * omt);
        float lad  = logf(dnum) - 2.0f * logf(denominator);

        ob[(size_t)(4 + c) * CSTRIDE] = inside ? outs : xv;
        ladsum += inside ? lad : 0.0f;
    }

    // ---- per-block logabsdet reduction (all pixels in block share batch b) --
    red[t] = ladsum;
    __syncthreads();
#pragma unroll
    for (int off = TPB / 2; off > 0; off >>= 1) {
        if (t < off) red[t] += red[t + off];
        __syncthreads();
    }
    if (t == 0) atomicAdd(&lad_out[b], red[0]);
}

extern "C" void kernel_launch(void* const* d_in, const int* in_sizes, int n_in,
                              void* d_out, int out_size, void* d_ws, size_t ws_size,
                              hipStream_t stream) {
    const float* x  = (const float*)d_in[0];   // (8, 8, 256, 256)
    const float* Wn = (const float*)d_in[1];   // (116, 4)
    const float* bn = (const float*)d_in[2];   // (116,)
    float* out = (float*)d_out;                // z (4194304) ++ logabsdet (8)
    float* lad = out + (size_t)8 * BSTRIDE;

    zero_lad_kernel<<<1, 32, 0, stream>>>(lad);
    spline_kernel<<<NBLOCKS, TPB, 0, stream>>>(x, Wn, bn, out, lad);
}